// SortAndSelectNeighbours_52785148067989
// MI455X (gfx1250) — compile-verified
//
#include <hip/hip_runtime.h>
#include <stdint.h>

typedef unsigned long long u64;

#define MCAND 64
#define KOUT  32
#define ROWS_PER_BLOCK 8
#define BIGF 1.0e9f
#define RADIUSF 0.5f

// Map float bits to a total-order-preserving unsigned (no NaNs in this data).
__device__ __forceinline__ unsigned f32_ordered(float f) {
    unsigned b = __float_as_uint(f);
    return b ^ ((unsigned)((int)b >> 31) | 0x80000000u);
}

__global__ __launch_bounds__(256)
void SortAndSelectNeighbours_kernel(const float* __restrict__ dist,
                                    const int*   __restrict__ nidx,
                                    float* __restrict__ outd,
                                    int*   __restrict__ outi,
                                    int V)
{
    __shared__ __align__(16) float shd[ROWS_PER_BLOCK][MCAND];
    __shared__ __align__(16) int   shi[ROWS_PER_BLOCK][MCAND];

    const int lane = threadIdx.x & 31;
    const int w    = threadIdx.x >> 5;          // wave id within block
    const int row  = blockIdx.x * ROWS_PER_BLOCK + w;
    if (row >= V) return;                       // wave-uniform branch

    // ---------- CDNA5 async global -> LDS staging (ASYNCcnt path) ----------
    // One B64 async op per array: 32 lanes x 8B = 256B = one full row.
    {
        unsigned ldsD = (unsigned)(uintptr_t)(&shd[w][0]) + (unsigned)lane * 8u;
        unsigned ldsI = (unsigned)(uintptr_t)(&shi[w][0]) + (unsigned)lane * 8u;
        unsigned goff = (unsigned)row * (MCAND * 4u) + (unsigned)lane * 8u;
        asm volatile("global_load_async_to_lds_b64 %0, %1, %2"
                     :: "v"(ldsD), "v"(goff), "s"(dist) : "memory");
        asm volatile("global_load_async_to_lds_b64 %0, %1, %2"
                     :: "v"(ldsI), "v"(goff), "s"(nidx) : "memory");
        asm volatile("s_wait_asynccnt 0" ::: "memory");
    }

    // Each lane owns columns 2*lane and 2*lane+1.
    float2 d2 = *(const float2*)&shd[w][2 * lane];
    int2   i2 = *(const int2*)  &shi[w][2 * lane];

    float s0 = (i2.x < 0) ? BIGF : d2.x;
    float s1 = (i2.y < 0) ? BIGF : d2.y;

    // Stable, distinct composite keys: (ordered float << 6) | column.
    u64 c0 = ((u64)f32_ordered(s0) << 6) | (unsigned)(2 * lane);
    u64 c1 = ((u64)f32_ordered(s1) << 6) | (unsigned)(2 * lane + 1);
    // Column 0 always sorts first (reference forces its key below min of the
    // rest; key 0 is strictly below every real composite key, incl. ties).
    if (lane == 0) c0 = 0ull;

    // ---------------- Bitonic sort, 64 keys, 2 per lane --------------------
    // Virtual position of c0 is 'lane', of c1 is 'lane + 32'.
    // Phase 1 (k = 2..32): build c0 ascending / c1 descending (both halves
    // progress in parallel, one CE per register per stage).
    #pragma unroll
    for (int k = 2; k <= 32; k <<= 1) {
        #pragma unroll
        for (int m = k >> 1; m > 0; m >>= 1) {
            u64 p0 = __shfl_xor(c0, m, 32);
            u64 p1 = __shfl_xor(c1, m, 32);
            // keep-max selector: ((i&m)!=0) XOR ((i&k)!=0), i = virtual pos
            bool sm0 = ((lane & m) != 0) != (( lane       & k) != 0);
            bool sm1 = ((lane & m) != 0) != (((lane + 32) & k) != 0);
            c0 = ((c0 < p0) != sm0) ? c0 : p0;   // keys are all distinct
            c1 = ((c1 < p1) != sm1) ? c1 : p1;
        }
    }

    // Phase 2 (k = 64, m = 32): split step of the bitonic merge. Lower half
    // (c0) now holds the 32 smallest keys as a bitonic sequence; upper half
    // (c1) holds the 32 largest and is never needed again -> prune it.
    {
        bool lt = c0 < c1;
        c0 = lt ? c0 : c1;                       // min -> position 'lane'
    }

    // Phase 3 (k = 64, m = 16..1): finish sorting only the lower half.
    // Direction is uniformly ascending; pairs never cross the halves.
    #pragma unroll
    for (int m = 16; m > 0; m >>= 1) {
        u64 p0 = __shfl_xor(c0, m, 32);
        bool sm0 = (lane & m) != 0;              // keep-max for upper partner
        c0 = ((c0 < p0) != sm0) ? c0 : p0;
    }

    // Sorted position 'lane' (0..31) now sits in c0: exactly the K kept
    // entries, in order. Gather original dist/idx by column (conflict-free:
    // 32 distinct columns -> 32 distinct LDS banks), apply radius mask,
    // store fully coalesced.
    int   col = (int)(c0 & 63u);
    float d   = shd[w][col];
    int   id  = shi[w][col];
    bool  far = d > RADIUSF;

    const size_t obase = (size_t)row * KOUT;
    outd[obase + lane] = far ? 0.0f : d;
    outi[obase + lane] = far ? -1   : id;
}

extern "C" void kernel_launch(void* const* d_in, const int* in_sizes, int n_in,
                              void* d_out, int out_size, void* d_ws, size_t ws_size,
                              hipStream_t stream) {
    const float* dist = (const float*)d_in[0];   // [V, 64] f32
    const int*   nidx = (const int*)  d_in[1];   // [V, 64] i32
    const int V = in_sizes[0] / MCAND;

    float* outd = (float*)d_out;                            // sdist [V, 32] f32
    int*   outi = (int*)((float*)d_out + (size_t)V * KOUT); // snidx [V, 32] i32

    const int blocks = (V + ROWS_PER_BLOCK - 1) / ROWS_PER_BLOCK;
    SortAndSelectNeighbours_kernel<<<blocks, 256, 0, stream>>>(dist, nidx, outd, outi, V);
}